// FastEquiformerV2ForceHead_1949915152411
// MI455X (gfx1250) — compile-verified
//
#include <hip/hip_runtime.h>
#include <hip/hip_bf16.h>

// ---------------------------------------------------------------------------
// FastEquiformerV2ForceHead on gfx1250 (CDNA5), fp32 WMMA (16x16x4) pipeline.
//
// Sizes: N=10000, E=100000, C=64, L2=9, H=8, DH=16, DV=16, EC=64, NRBF=128.
//
// Folding (keeps everything small / L2-resident; wigner is the only HBM stream):
//   U[c][h]      = sum_v W_val[c][16h+v] * W_out[16h+v]           (128x8)
//   P[n][j][:]   = ne[n][j][:] @ W_msg[0:64 ,:]   \  one dense GEMM
//   Q[n][j][:]   = ne[n][j][:] @ W_msg[64:128,:]  /  (90000x64)@(64x256)
//   msg[e]       = wigner[e] @ (P[src]+Q[dst])        (9x9 @ 9x128)
//   s[e]         = (msg[e]*gate[e]) @ U               (9x128 @ 128x8)
//   g[e][i][h]   = sum_j wigner[e][j][i+1] * s[e][j][h]   (i=0..2)
//   forces[n][i] = sum_h (softmax-weighted segment sums) / den
// ---------------------------------------------------------------------------

#define NN      10000
#define EE      100000
#define LL2     9
#define HH      8
#define WAVES   4          // waves per block in WMMA kernels (128 threads)

typedef float v2f __attribute__((ext_vector_type(2)));
typedef float v8f __attribute__((ext_vector_type(8)));

#define V8ZERO {0.f,0.f,0.f,0.f,0.f,0.f,0.f,0.f}

static __device__ __forceinline__ v8f wmma4(v2f a, v2f b, v8f c) {
  // V_WMMA_F32_16X16X4_F32 : D(16x16,f32) = A(16x4,f32) x B(4x16,f32) + C
  return __builtin_amdgcn_wmma_f32_16x16x4_f32(
      false, a, false, b, (short)0, c, false, false);
}

#define SWZ(x, off) \
  __int_as_float(__builtin_amdgcn_ds_swizzle(__float_as_int(x), (off)))

// All per-wave LDS handoffs: CDNA5 LDS ops are in-order within a wave, so a
// compiler reordering fence is sufficient (no cross-wave sharing).
#define WAVE_LDS_FENCE() asm volatile("" ::: "memory")

#if __has_builtin(__builtin_amdgcn_global_load_async_to_lds_b32) && \
    __has_builtin(__builtin_amdgcn_s_wait_asynccnt)
#define HAVE_ASYNC_LDS 1
typedef __attribute__((address_space(1))) int* gas1_i32;
typedef __attribute__((address_space(3))) int* las3_i32;
#else
#define HAVE_ASYNC_LDS 0
#endif

static __device__ __forceinline__ float silu_f(float x) {
  return x / (1.0f + __expf(-x));
}

// ---------------------------------------------------------------------------
// K0: init accumulators + build U (128x8)
// ---------------------------------------------------------------------------
__global__ void k_init(const float* __restrict__ Wval,
                       const float* __restrict__ Wout,
                       unsigned* __restrict__ mkeys,
                       float* __restrict__ den,
                       float* __restrict__ fnum,
                       float* __restrict__ U) {
  int t = blockIdx.x * blockDim.x + threadIdx.x;
  if (t < NN * HH) { mkeys[t] = 0u; den[t] = 0.0f; }
  if (t < NN * 3 * HH) fnum[t] = 0.0f;
  if (t < 128 * HH) {
    int c = t >> 3, h = t & 7;
    float s = 0.0f;
    #pragma unroll
    for (int v = 0; v < 16; ++v)
      s += Wval[c * 128 + h * 16 + v] * Wout[h * 16 + v];
    U[t] = s;
  }
}

// ---------------------------------------------------------------------------
// K1: node precompute  Xpq[(n*9+j)][0:128]=P, [128:256]=Q
//     dense GEMM (90000 x 64) @ (64 x 256), one 16-row tile per wave.
//     B fragments preloaded per n-tile -> one wait, 16 back-to-back WMMAs.
// ---------------------------------------------------------------------------
__global__ void __launch_bounds__(WAVES * 32)
k_nodes(const float* __restrict__ ne,
        const float* __restrict__ Wmsg,
        float* __restrict__ Xpq) {
  const int L    = threadIdx.x & 31;
  const int w    = threadIdx.x >> 5;
  const int half = L >> 4, l16 = L & 15;
  const int MT   = (NN * LL2) / 16;                 // 5625 (exact)
  int tile = blockIdx.x * WAVES + w;
  if (tile >= MT) tile = MT - 1;                    // idempotent redo
  const long r0 = (long)tile * 16;

  // A fragments of the 16-row slab (row = l16, k = 4kk + 2*half + v)
  v2f ea[16];
  #pragma unroll
  for (int kk = 0; kk < 16; ++kk)
    ea[kk] = *(const v2f*)&ne[(r0 + l16) * 64 + 4 * kk + 2 * half];

  for (int nt = 0; nt < 16; ++nt) {
    const int t = nt * 16 + l16;                    // 0..255 (t<128 uniform/nt)
    const float* bbase = (t < 128) ? (Wmsg + t) : (Wmsg + 64 * 128 + (t - 128));
    v2f bf[16];
    #pragma unroll
    for (int kk = 0; kk < 16; ++kk) {
      const int k0 = 4 * kk + 2 * half;
      bf[kk].x = bbase[(k0    ) * 128];
      bf[kk].y = bbase[(k0 + 1) * 128];
    }
    v8f c8 = V8ZERO;
    #pragma unroll
    for (int kk = 0; kk < 16; ++kk) c8 = wmma4(ea[kk], bf[kk], c8);
    #pragma unroll
    for (int r = 0; r < 8; ++r)
      Xpq[(r0 + r + 8 * half) * 256 + t] = c8[r];
  }
}

// ---------------------------------------------------------------------------
// K2: edge pass A — per wave: 16 edges.
// ---------------------------------------------------------------------------
__global__ void __launch_bounds__(WAVES * 32)
k_edge_a(const int*   __restrict__ az,
         const float* __restrict__ dist,
         const int*   __restrict__ ei,
         const float* __restrict__ wig,
         const float* __restrict__ aemb,
         const float* __restrict__ Wrbf,
         const float* __restrict__ Wgate,
         const float* __restrict__ alpha,
         const float* __restrict__ Xpq,
         const float* __restrict__ U,
         float*    __restrict__ logitsW,
         float*    __restrict__ gW,
         unsigned* __restrict__ mkeys) {
  __shared__ float sh_wrbf[128 * 64];         // 32 KB, block-shared W_rbf
  __shared__ float sh_gate[WAVES][16][128];   // 32 KB
  __shared__ float sh_es  [WAVES][16][64];    // 16 KB
  __shared__ float sh_msg [WAVES][16][128];   // 32 KB (aliased as atom-emb early)

  const int L    = threadIdx.x & 31;
  const int w    = threadIdx.x >> 5;
  const int half = L >> 4, l16 = L & 15;
  const int ET   = EE / 16;                          // 6250 (exact)
  int tile = blockIdx.x * WAVES + w;
  if (tile >= ET) tile = ET - 1;                     // idempotent redo
  const int e0 = tile * 16;

  // ---- stage W_rbf into LDS (async copy if toolchain exposes it) ----
#if HAVE_ASYNC_LDS
  for (int t = threadIdx.x; t < 128 * 64; t += WAVES * 32)
    __builtin_amdgcn_global_load_async_to_lds_b32(
        (gas1_i32)(void*)(Wrbf + t), (las3_i32)(void*)&sh_wrbf[t], 0, 0);
  __builtin_amdgcn_s_wait_asynccnt(0);
#else
  for (int t = threadIdx.x; t < 128 * 64; t += WAVES * 32)
    sh_wrbf[t] = Wrbf[t];
#endif

  // ---- cooperative atom-embedding staging (alias sh_msg; wave-local) ----
  float* sh_ae = &sh_msg[w][0][0];                   // 16 x 64
  for (int t = L; t < 16 * 64; t += 32) {
    const int m = t >> 6, c = t & 63;
    const int e = e0 + m;
    const int s = ei[e], d = ei[EE + e];
    sh_ae[t] = aemb[az[s] * 64 + c] + aemb[az[d] * 64 + c];
  }
  __syncthreads();                                   // covers sh_wrbf (block-wide)

  // ---- per-lane RBF A-fragments: edge m=l16, k=4kk+2*half+v ----
  const float dm = dist[e0 + l16];
  v2f ra[32];
  #pragma unroll
  for (int kk = 0; kk < 32; ++kk) {
    const int k0 = 4 * kk + 2 * half;
    const float t0 = dm - (float)k0 * (1.0f / 127.0f);
    const float t1 = dm - (float)(k0 + 1) * (1.0f / 127.0f);
    ra[kk].x = __expf(-t0 * t0 * 8192.0f);           // 1/(2w^2), w=1/128
    ra[kk].y = __expf(-t1 * t1 * 8192.0f);
  }

  // ---- stage 1: edge_scalar = silu(rbf @ W_rbf + ae)  (16 x 64) ----
  for (int nt = 0; nt < 4; ++nt) {
    v8f c8 = V8ZERO;
    #pragma unroll
    for (int cc = 0; cc < 2; ++cc) {                 // chunks of 16 k-steps
      v2f bf[16];
      #pragma unroll
      for (int kk = 0; kk < 16; ++kk) {
        const int k0 = 4 * (cc * 16 + kk) + 2 * half;
        bf[kk].x = sh_wrbf[(k0    ) * 64 + nt * 16 + l16];
        bf[kk].y = sh_wrbf[(k0 + 1) * 64 + nt * 16 + l16];
      }
      #pragma unroll
      for (int kk = 0; kk < 16; ++kk)
        c8 = wmma4(ra[cc * 16 + kk], bf[kk], c8);
    }
    #pragma unroll
    for (int r = 0; r < 8; ++r) {
      const int m = r + 8 * half, c = nt * 16 + l16;
      sh_es[w][m][c] = silu_f(c8[r] + sh_ae[m * 64 + c]);
    }
  }
  WAVE_LDS_FENCE();

  // ---- stage 2: gate = silu(es @ W_gate)  (16 x 128) ----
  v2f ea[16];
  #pragma unroll
  for (int kk = 0; kk < 16; ++kk)
    ea[kk] = *(const v2f*)&sh_es[w][l16][4 * kk + 2 * half];
  for (int nt = 0; nt < 8; ++nt) {
    v2f bf[16];
    #pragma unroll
    for (int kk = 0; kk < 16; ++kk) {
      const int k0 = 4 * kk + 2 * half;
      bf[kk].x = Wgate[(k0    ) * 128 + nt * 16 + l16];
      bf[kk].y = Wgate[(k0 + 1) * 128 + nt * 16 + l16];
    }
    v8f c8 = V8ZERO;
    #pragma unroll
    for (int kk = 0; kk < 16; ++kk) c8 = wmma4(ea[kk], bf[kk], c8);
    #pragma unroll
    for (int r = 0; r < 8; ++r)
      sh_gate[w][r + 8 * half][nt * 16 + l16] = silu_f(c8[r]);
  }
  WAVE_LDS_FENCE();

  // per-lane alpha row (h = nt, d = l16)
  float aw[8];
  #pragma unroll
  for (int nt = 0; nt < 8; ++nt) aw[nt] = alpha[nt * 16 + l16];

  // U B-fragments are invariant across the 16 edges: hoist into registers.
  v2f ub[32];
  #pragma unroll
  for (int kk = 0; kk < 32; ++kk) {
    const int k0 = 4 * kk + 2 * half;
    ub[kk].x = (l16 < 8) ? U[(k0    ) * 8 + l16] : 0.0f;
    ub[kk].y = (l16 < 8) ? U[(k0 + 1) * 8 + l16] : 0.0f;
  }

  // ---- stage 3: per-edge equivariant pipeline ----
  for (int em = 0; em < 16; ++em) {
    const int  e    = e0 + em;
    __builtin_prefetch(&wig[(long)(e + 1) * 81], 0, 0);   // stream next tile
    const int  srci = ei[e];
    const int  dsti = ei[EE + e];
    const long sB   = (long)srci * 9;
    const long dB   = (long)dsti * 9;

    // A = wigner[e] (rows j=l16 pad16, cols k pad12, zero k>=9)
    v2f wa[3];
    #pragma unroll
    for (int kk = 0; kk < 3; ++kk)
      #pragma unroll
      for (int v = 0; v < 2; ++v) {
        const int  k     = 4 * kk + 2 * half + v;
        const bool valid = (l16 < 9) && (k < 9);
        const long idx   = valid ? ((long)e * 81 + l16 * 9 + k) : 0;
        const float x    = wig[idx];
        wa[kk][v] = valid ? x : 0.0f;
      }

    // msg = wigner @ (P[src]+Q[dst]), gated   (D-layout fragments)
    v8f mfr[8];
    #pragma unroll
    for (int nt = 0; nt < 8; ++nt) {
      v2f bfr[3];
      #pragma unroll
      for (int kk = 0; kk < 3; ++kk)
        #pragma unroll
        for (int v = 0; v < 2; ++v) {
          const int k = 4 * kk + 2 * half + v;
          float rv = 0.0f;
          if (k < 9) {
            rv = Xpq[(sB + k) * 256 + nt * 16 + l16]
               + Xpq[(dB + k) * 256 + 128 + nt * 16 + l16];
          }
          bfr[kk][v] = rv;
        }
      v8f c8 = V8ZERO;
      #pragma unroll
      for (int kk = 0; kk < 3; ++kk) c8 = wmma4(wa[kk], bfr[kk], c8);
      const float gv = sh_gate[w][em][nt * 16 + l16];
      #pragma unroll
      for (int r = 0; r < 8; ++r) c8[r] *= gv;
      mfr[nt] = c8;
    }

    // logits[h] = sum_d lrelu(msg[0, 16h+d], 0.2) * alpha[h][d]
    #pragma unroll
    for (int nt = 0; nt < 8; ++nt) {
      float x = mfr[nt][0];                 // row j=0 lives in half-0, reg 0
      x = (x > 0.0f) ? x : 0.2f * x;
      x *= aw[nt];
      x += SWZ(x, 0x041F);                  // butterfly over lanes 0..15
      x += SWZ(x, 0x081F);
      x += SWZ(x, 0x101F);
      x += SWZ(x, 0x201F);
      if (L == 0) {
        logitsW[(long)e * 8 + nt] = x;
        const unsigned bits = __float_as_uint(x);
        const unsigned key  = (bits & 0x80000000u) ? ~bits
                                                   : (bits | 0x80000000u);
        atomicMax(&mkeys[dsti * 8 + nt], key);
      }
    }

    // bounce msg (D-layout) through LDS to build A-fragments for s = msg @ U
    #pragma unroll
    for (int nt = 0; nt < 8; ++nt)
      #pragma unroll
      for (int r = 0; r < 8; ++r)
        sh_msg[w][r + 8 * half][nt * 16 + l16] = mfr[nt][r];
    WAVE_LDS_FENCE();

    v8f s8 = V8ZERO;
    #pragma unroll
    for (int cc = 0; cc < 4; ++cc) {        // chunks of 8 k-steps
      v2f af[8];
      #pragma unroll
      for (int kk = 0; kk < 8; ++kk)
        af[kk] = *(const v2f*)&sh_msg[w][l16][4 * (cc * 8 + kk) + 2 * half];
      #pragma unroll
      for (int kk = 0; kk < 8; ++kk)
        s8 = wmma4(af[kk], ub[cc * 8 + kk], s8);
    }
    WAVE_LDS_FENCE();

    // g[i][h] = sum_j wigner[e][j][i+1] * s[j][h]
    float gacc[3];
    #pragma unroll
    for (int i = 0; i < 3; ++i) {
      float part = 0.0f;
      #pragma unroll
      for (int r = 0; r < 8; ++r) {
        const int  j     = r + 8 * half;
        const bool valid = (j < 9);
        const long idx   = valid ? ((long)e * 81 + j * 9 + (i + 1)) : 0;
        const float wv   = wig[idx];
        part += (valid ? wv : 0.0f) * s8[r];
      }
      part += SWZ(part, 0x401F);            // add the other half-wave
      gacc[i] = part;
    }
    if (L < 8) {
      #pragma unroll
      for (int i = 0; i < 3; ++i)
        gW[(long)e * 24 + i * 8 + L] = gacc[i];
    }
  }
}

// ---------------------------------------------------------------------------
// K3: edge pass B — softmax numerator/denominator scatter (one thread/(e,h))
// ---------------------------------------------------------------------------
__global__ void k_edge_b(const int*      __restrict__ ei,
                         const unsigned* __restrict__ mkeys,
                         const float*    __restrict__ logitsW,
                         const float*    __restrict__ gW,
                         float* __restrict__ den,
                         float* __restrict__ fnum) {
  const long t = (long)blockIdx.x * blockDim.x + threadIdx.x;
  if (t >= (long)EE * HH) return;
  const int e = (int)(t >> 3), h = (int)(t & 7);
  const int dst = ei[EE + e];
  const unsigned key = mkeys[dst * 8 + h];
  const float m = (key & 0x80000000u) ? __uint_as_float(key & 0x7FFFFFFFu)
                                      : __uint_as_float(~key);
  const float ex = __expf(logitsW[(long)e * 8 + h] - m);
  atomicAdd(&den[dst * 8 + h], ex);
  #pragma unroll
  for (int i = 0; i < 3; ++i)
    atomicAdd(&fnum[dst * 24 + i * 8 + h], ex * gW[(long)e * 24 + i * 8 + h]);
}

// ---------------------------------------------------------------------------
// K4: finalize forces[n][i] = sum_h fnum/(den+1e-9)
// ---------------------------------------------------------------------------
__global__ void k_final(const float* __restrict__ den,
                        const float* __restrict__ fnum,
                        float* __restrict__ out) {
  const int t = blockIdx.x * blockDim.x + threadIdx.x;
  if (t >= NN * 3) return;
  const int n = t / 3, i = t % 3;
  float acc = 0.0f;
  #pragma unroll
  for (int h = 0; h < 8; ++h)
    acc += fnum[n * 24 + i * 8 + h] / (den[n * 8 + h] + 1e-9f);
  out[t] = acc;
}

// ---------------------------------------------------------------------------
extern "C" void kernel_launch(void* const* d_in, const int* in_sizes, int n_in,
                              void* d_out, int out_size, void* d_ws,
                              size_t ws_size, hipStream_t stream) {
  const float* ne    = (const float*)d_in[0];
  const int*   az    = (const int*)  d_in[1];
  const float* dist  = (const float*)d_in[2];
  const int*   ei    = (const int*)  d_in[3];
  const float* wig   = (const float*)d_in[4];
  const float* aemb  = (const float*)d_in[5];
  const float* Wrbf  = (const float*)d_in[6];
  const float* Wgate = (const float*)d_in[7];
  const float* Wmsg  = (const float*)d_in[8];
  const float* alpha = (const float*)d_in[9];
  const float* Wval  = (const float*)d_in[10];
  const float* Wout  = (const float*)d_in[11];
  float* out = (float*)d_out;

  // workspace layout (floats)
  float*    ws      = (float*)d_ws;
  float*    Xpq     = ws;                               // 90000*256
  float*    U       = Xpq + (long)NN * LL2 * 256;       // 1024
  unsigned* mkeys   = (unsigned*)(U + 128 * HH);        // N*8
  float*    den     = (float*)(mkeys + NN * HH);        // N*8
  float*    fnum    = den + NN * HH;                    // N*24
  float*    logitsW = fnum + NN * 3 * HH;               // E*8
  float*    gW      = logitsW + (long)EE * HH;          // E*24

  {
    const int tot = NN * 3 * HH;                        // covers all init work
    k_init<<<(tot + 255) / 256, 256, 0, stream>>>(Wval, Wout, mkeys, den,
                                                  fnum, U);
  }
  {
    const int tiles = (NN * LL2) / 16;                  // 5625
    const int blocks = (tiles + WAVES - 1) / WAVES;
    k_nodes<<<blocks, WAVES * 32, 0, stream>>>(ne, Wmsg, Xpq);
  }
  {
    const int tiles = EE / 16;                          // 6250
    const int blocks = (tiles + WAVES - 1) / WAVES;
    k_edge_a<<<blocks, WAVES * 32, 0, stream>>>(az, dist, ei, wig, aemb, Wrbf,
                                                Wgate, alpha, Xpq, U, logitsW,
                                                gW, mkeys);
  }
  {
    const long tot = (long)EE * HH;
    k_edge_b<<<(int)((tot + 255) / 256), 256, 0, stream>>>(ei, mkeys, logitsW,
                                                           gW, den, fnum);
  }
  {
    const int tot = NN * 3;
    k_final<<<(tot + 255) / 256, 256, 0, stream>>>(den, fnum, out);
  }
}